// TopkAttention_3865470566724
// MI455X (gfx1250) — compile-verified
//
#include <hip/hip_runtime.h>
#include <hip/hip_bf16.h>
#include <math.h>

// ---------------------------------------------------------------------------
// TopkAttention for MI455X (gfx1250, wave32, WMMA).
// All GEMM stages use V_WMMA_F32_16X16X4_F32 (full fp32 matrix path):
// the pipeline is memory-bound (~2 GB HBM traffic vs ~90 GFLOP), so fp32
// WMMA costs nothing at the roofline and keeps exact-precision top-k
// thresholding safe.
// ---------------------------------------------------------------------------

#define B_     2
#define DIM_   384
#define QKV_   1152
#define HEADS_ 8
#define C_     48
#define H_     192
#define W_     192
#define N_     36864          // H_*W_

typedef __attribute__((ext_vector_type(2))) float v2f;
typedef __attribute__((ext_vector_type(8))) float v8f;

__device__ __forceinline__ v8f wmma_f32(v2f a, v2f b, v8f c) {
  // D = A(16x4,f32) * B(4x16,f32) + C(16x16,f32)
  return __builtin_amdgcn_wmma_f32_16x16x4_f32(false, a, false, b, (short)0, c,
                                               false, false);
}

// ---------------------------------------------------------------------------
// Kernel 1/7: Y[b][m][n] = sum_k W[m][k] * X[b][k][n]
// block = 256 threads (8 waves), block tile 64(M) x 32(N), K chunks of 16 via
// LDS (padded to avoid bank conflicts). Each wave owns one 16x16 WMMA tile.
// ---------------------------------------------------------------------------
__global__ __launch_bounds__(256) void k_conv1x1(const float* __restrict__ W,
                                                 const float* __restrict__ X,
                                                 float* __restrict__ Y,
                                                 int M, int K, int N) {
  __shared__ float Wt[64][17];
  __shared__ float Xt[16][33];

  const int b    = blockIdx.z;
  const int mBlk = blockIdx.y * 64;
  const int nBlk = blockIdx.x * 32;
  const float* Xb = X + (size_t)b * K * N;
  float*       Yb = Y + (size_t)b * M * N;

  const int tid  = threadIdx.x;
  const int lane = tid & 31;
  const int wave = tid >> 5;
  const int r    = lane & 15;
  const int half = lane >> 4;
  const int mW   = (wave & 3) * 16;   // wave M sub-tile within block
  const int nW   = (wave >> 2) * 16;  // wave N sub-tile within block

  v8f acc = {0.f, 0.f, 0.f, 0.f, 0.f, 0.f, 0.f, 0.f};

  for (int k0 = 0; k0 < K; k0 += 16) {
    // stage W 64x16 (4 consecutive k per thread, float4)
    {
      int idx = tid * 4;            // 0..1023
      int mm = idx >> 4, kk = idx & 15;
      float4 w4 = *(const float4*)(W + (size_t)(mBlk + mm) * K + k0 + kk);
      Wt[mm][kk + 0] = w4.x; Wt[mm][kk + 1] = w4.y;
      Wt[mm][kk + 2] = w4.z; Wt[mm][kk + 3] = w4.w;
    }
    // stage X 16x32 (2 consecutive n per thread, float2)
    {
      int idx = tid * 2;            // 0..511
      int kk = idx >> 5, nn = idx & 31;
      float2 x2 = *(const float2*)(Xb + (size_t)(k0 + kk) * N + nBlk + nn);
      Xt[kk][nn + 0] = x2.x; Xt[kk][nn + 1] = x2.y;
    }
    __syncthreads();
#pragma unroll
    for (int k4 = 0; k4 < 16; k4 += 4) {
      v2f a, bb;
      // A 16x4: lane=M row, VGPR0/1 hold K = {k4+2*half, k4+2*half+1}
      a.x  = Wt[mW + r][k4 + 2 * half];
      a.y  = Wt[mW + r][k4 + 2 * half + 1];
      // B 4x16: VGPR0/1 hold K rows {2*half, 2*half+1}, lane=N col
      bb.x = Xt[k4 + 2 * half][nW + r];
      bb.y = Xt[k4 + 2 * half + 1][nW + r];
      acc = wmma_f32(a, bb, acc);
    }
    __syncthreads();
  }
  // C/D 16x16: VGPR g -> M = g + 8*half, lane -> N = r
#pragma unroll
  for (int g = 0; g < 8; ++g) {
    int m = mBlk + mW + g + 8 * half;
    int n = nBlk + nW + r;
    Yb[(size_t)m * N + n] = acc[g];
  }
}

// ---------------------------------------------------------------------------
// Kernel 2: depthwise 3x3, pad 1 (pure bandwidth).
// ---------------------------------------------------------------------------
__global__ __launch_bounds__(256) void k_dwconv(const float* __restrict__ src,
                                                const float* __restrict__ wd,
                                                float* __restrict__ dst) {
  size_t gid = (size_t)blockIdx.x * blockDim.x + threadIdx.x;
  int j  = (int)(gid % W_);
  size_t t = gid / W_;
  int i  = (int)(t % H_);
  size_t bc = t / H_;               // b*QKV_+ch
  int ch = (int)(bc % QKV_);
  const float* s  = src + bc * (size_t)N_;
  const float* wp = wd + ch * 9;
  float acc = 0.f;
#pragma unroll
  for (int di = 0; di < 3; ++di) {
    int ii = i + di - 1;
    if (ii < 0 || ii >= H_) continue;
#pragma unroll
    for (int dj = 0; dj < 3; ++dj) {
      int jj = j + dj - 1;
      if (jj < 0 || jj >= W_) continue;
      acc += wp[di * 3 + dj] * s[(size_t)ii * W_ + jj];
    }
  }
  dst[gid] = acc;
}

// ---------------------------------------------------------------------------
// Kernel 3: reciprocal L2 norms of q (ch 0..383) and k (ch 384..767) rows.
// rnorm[b][ch] = 1 / max(||row||, 1e-12)
// ---------------------------------------------------------------------------
__global__ __launch_bounds__(256) void k_rownorm(const float* __restrict__ qkv,
                                                 float* __restrict__ rnorm) {
  const int ch = blockIdx.x;        // 0..767
  const int b  = blockIdx.y;
  const float* row = qkv + ((size_t)b * QKV_ + ch) * N_;
  float s = 0.f;
  for (int n = threadIdx.x; n < N_; n += 256) {
    float v = row[n];
    s += v * v;
  }
  __shared__ float red[256];
  red[threadIdx.x] = s;
  __syncthreads();
  for (int off = 128; off > 0; off >>= 1) {
    if (threadIdx.x < off) red[threadIdx.x] += red[threadIdx.x + off];
    __syncthreads();
  }
  if (threadIdx.x == 0)
    rnorm[b * 768 + ch] = 1.0f / fmaxf(sqrtf(red[0]), 1e-12f);
}

// ---------------------------------------------------------------------------
// Kernel 4: raw attn[b][h][c][d] = sum_n q[c][n]*k[d][n]   (scaling later)
// 288 threads = 9 waves = full 3x3 grid of 16x16 tiles of the 48x48 output.
// K (=36864) split across blockIdx.x, partials combined with atomicAdd.
// ---------------------------------------------------------------------------
__global__ __launch_bounds__(288) void k_attn(const float* __restrict__ qkv,
                                              float* __restrict__ attn,
                                              int klen) {
  __shared__ float qs[48][33];
  __shared__ float ks[48][33];

  const int bh = blockIdx.y;        // 0..15
  const int b  = bh >> 3;
  const int h  = bh & 7;
  const int kBeg = blockIdx.x * klen;

  const float* qrow = qkv + (size_t)b * QKV_ * N_ + (size_t)(h * C_) * N_;
  const float* krow = qkv + (size_t)b * QKV_ * N_ + (size_t)(DIM_ + h * C_) * N_;

  const int tid  = threadIdx.x;
  const int lane = tid & 31;
  const int wave = tid >> 5;        // 0..8
  const int r    = lane & 15;
  const int half = lane >> 4;
  const int cW   = (wave / 3) * 16;
  const int dW   = (wave % 3) * 16;

  v8f acc = {0.f, 0.f, 0.f, 0.f, 0.f, 0.f, 0.f, 0.f};

  for (int k0 = kBeg; k0 < kBeg + klen; k0 += 32) {
    for (int idx = tid; idx < 1536; idx += 288) {
      int row = idx >> 5, kk = idx & 31;
      qs[row][kk] = qrow[(size_t)row * N_ + k0 + kk];
      ks[row][kk] = krow[(size_t)row * N_ + k0 + kk];
    }
    __syncthreads();
#pragma unroll
    for (int k4 = 0; k4 < 32; k4 += 4) {
      v2f a, bb;
      a.x  = qs[cW + r][k4 + 2 * half];
      a.y  = qs[cW + r][k4 + 2 * half + 1];
      bb.x = ks[dW + r][k4 + 2 * half];
      bb.y = ks[dW + r][k4 + 2 * half + 1];
      acc = wmma_f32(a, bb, acc);
    }
    __syncthreads();
  }
#pragma unroll
  for (int g = 0; g < 8; ++g) {
    int c = cW + g + 8 * half;
    int d = dW + r;
    atomicAdd(&attn[((size_t)bh * C_ + c) * C_ + d], acc[g]);
  }
}

// ---------------------------------------------------------------------------
// Kernel 5: scale by rq*rk*temperature, then 4 exact top-k softmaxes mixed
// with a1..a4 into one matrix P (written back in place over attn).
// thr_k = min{ x_j : #(x_i > x_j) < k }  == k-th largest (lax.top_k semantics)
// ---------------------------------------------------------------------------
__global__ __launch_bounds__(64) void k_topkmix(float* __restrict__ attnP,
                                                const float* __restrict__ rnorm,
                                                const float* __restrict__ temp,
                                                const float* __restrict__ a1,
                                                const float* __restrict__ a2,
                                                const float* __restrict__ a3,
                                                const float* __restrict__ a4) {
  const int h = blockIdx.x;
  const int b = blockIdx.y;
  __shared__ float As[48][48];
  __shared__ float aw[4];

  const float tmp = temp[h];
  const float* rq = rnorm + b * 768 + h * C_;
  const float* rk = rnorm + b * 768 + DIM_ + h * C_;
  float* rowP = attnP + (size_t)(b * HEADS_ + h) * C_ * C_;

  for (int idx = threadIdx.x; idx < C_ * C_; idx += 64) {
    int c = idx / C_, d = idx % C_;
    As[c][d] = rowP[idx] * rq[c] * rk[d] * tmp;
  }
  if (threadIdx.x == 0) { aw[0] = a1[0]; aw[1] = a2[0]; aw[2] = a3[0]; aw[3] = a4[0]; }
  __syncthreads();

  if (threadIdx.x < C_) {
    const int c = threadIdx.x;
    const int kk[4] = {C_ / 2, C_ * 2 / 3, C_ * 3 / 4, C_ * 4 / 5}; // 24,32,36,38
    float thr[4] = {1e30f, 1e30f, 1e30f, 1e30f};
    float mx = -1e30f;
    for (int j = 0; j < C_; ++j) {
      float xj = As[c][j];
      mx = fmaxf(mx, xj);
      int cnt = 0;
      for (int i = 0; i < C_; ++i) cnt += (As[c][i] > xj) ? 1 : 0;
#pragma unroll
      for (int t = 0; t < 4; ++t)
        if (cnt < kk[t]) thr[t] = fminf(thr[t], xj);
    }
    float sum[4] = {0.f, 0.f, 0.f, 0.f};
    for (int j = 0; j < C_; ++j) {
      float e = expf(As[c][j] - mx);
#pragma unroll
      for (int t = 0; t < 4; ++t)
        if (As[c][j] >= thr[t]) sum[t] += e;
    }
    float inv[4];
#pragma unroll
    for (int t = 0; t < 4; ++t) inv[t] = aw[t] / sum[t];
    for (int j = 0; j < C_; ++j) {
      float e = expf(As[c][j] - mx);
      float p = 0.f;
#pragma unroll
      for (int t = 0; t < 4; ++t)
        if (As[c][j] >= thr[t]) p += e * inv[t];
      rowP[c * C_ + j] = p;
    }
  }
}

// ---------------------------------------------------------------------------
// Kernel 6: out[b][h][c][n] = sum_d P[c][d] * v[d][n]
// P (48x48) cached in LDS; v streamed once per block; 3 acc tiles per wave.
// ---------------------------------------------------------------------------
__global__ __launch_bounds__(256) void k_attnout(const float* __restrict__ qkv,
                                                 const float* __restrict__ P,
                                                 float* __restrict__ out) {
  __shared__ float Ps[48][49];
  const int bh   = blockIdx.y;
  const int b    = bh >> 3;
  const int h    = bh & 7;
  const int nBlk = blockIdx.x * 128;

  const float* vrow = qkv + (size_t)b * QKV_ * N_ + (size_t)(2 * DIM_ + h * C_) * N_;
  float*       orow = out + (size_t)b * DIM_ * N_ + (size_t)(h * C_) * N_;

  for (int idx = threadIdx.x; idx < C_ * C_; idx += 256)
    Ps[idx / C_][idx % C_] = P[(size_t)bh * C_ * C_ + idx];
  __syncthreads();

  const int tid  = threadIdx.x;
  const int lane = tid & 31;
  const int wave = tid >> 5;
  const int r    = lane & 15;
  const int half = lane >> 4;
  const int n    = nBlk + wave * 16 + r;

  v8f acc0 = {0.f, 0.f, 0.f, 0.f, 0.f, 0.f, 0.f, 0.f};
  v8f acc1 = acc0, acc2 = acc0;

#pragma unroll
  for (int kk = 0; kk < C_; kk += 4) {
    v2f bb;
    bb.x = vrow[(size_t)(kk + 2 * half) * N_ + n];
    bb.y = vrow[(size_t)(kk + 2 * half + 1) * N_ + n];
    v2f a0, a1v, a2v;
    a0.x  = Ps[0  + r][kk + 2 * half];  a0.y  = Ps[0  + r][kk + 2 * half + 1];
    a1v.x = Ps[16 + r][kk + 2 * half];  a1v.y = Ps[16 + r][kk + 2 * half + 1];
    a2v.x = Ps[32 + r][kk + 2 * half];  a2v.y = Ps[32 + r][kk + 2 * half + 1];
    acc0 = wmma_f32(a0,  bb, acc0);
    acc1 = wmma_f32(a1v, bb, acc1);
    acc2 = wmma_f32(a2v, bb, acc2);
  }
#pragma unroll
  for (int g = 0; g < 8; ++g) {
    int m = g + 8 * half;
    orow[(size_t)(m +  0) * N_ + n] = acc0[g];
    orow[(size_t)(m + 16) * N_ + n] = acc1[g];
    orow[(size_t)(m + 32) * N_ + n] = acc2[g];
  }
}

// ---------------------------------------------------------------------------
// kernel_launch
// workspace layout (floats):
//   qkv_pre  : B*QKV*N   (339.7 MB)  -- reused as attn_out after dwconv
//   qkv_post : B*QKV*N   (339.7 MB)
//   rnorm    : B*768
//   attn/P   : B*8*48*48
// ---------------------------------------------------------------------------
extern "C" void kernel_launch(void* const* d_in, const int* in_sizes, int n_in,
                              void* d_out, int out_size, void* d_ws, size_t ws_size,
                              hipStream_t stream) {
  const float* x      = (const float*)d_in[0];
  const float* w_qkv  = (const float*)d_in[1];
  const float* w_dw   = (const float*)d_in[2];
  const float* w_proj = (const float*)d_in[3];
  const float* temp   = (const float*)d_in[4];
  const float* a1     = (const float*)d_in[5];
  const float* a2     = (const float*)d_in[6];
  const float* a3     = (const float*)d_in[7];
  const float* a4     = (const float*)d_in[8];
  float* outp = (float*)d_out;

  const size_t QKV_ELEMS = (size_t)B_ * QKV_ * N_;
  float* qkv_pre  = (float*)d_ws;
  float* qkv_post = qkv_pre + QKV_ELEMS;
  float* rnorm    = qkv_post + QKV_ELEMS;
  float* attn     = rnorm + (size_t)B_ * 768;     // also P after k_topkmix
  float* attn_out = qkv_pre;                      // alias (pre is dead)

  // 1) qkv = w_qkv @ x       (M=1152, K=384, N=36864)
  k_conv1x1<<<dim3(N_ / 32, QKV_ / 64, B_), 256, 0, stream>>>(
      w_qkv, x, qkv_pre, QKV_, DIM_, N_);

  // 2) depthwise 3x3
  {
    size_t total = (size_t)B_ * QKV_ * N_;
    k_dwconv<<<(unsigned)(total / 256), 256, 0, stream>>>(qkv_pre, w_dw, qkv_post);
  }

  // 3) reciprocal L2 norms of q,k rows
  k_rownorm<<<dim3(768, B_), 256, 0, stream>>>(qkv_post, rnorm);

  // 4) raw attention dot products (split-K, atomics)
  hipMemsetAsync(attn, 0, (size_t)B_ * HEADS_ * C_ * C_ * sizeof(float), stream);
  k_attn<<<dim3(8, B_ * HEADS_), 288, 0, stream>>>(qkv_post, attn, N_ / 8);

  // 5) scale + 4x top-k softmax mixed into P (in place)
  k_topkmix<<<dim3(HEADS_, B_), 64, 0, stream>>>(attn, rnorm, temp, a1, a2, a3, a4);

  // 6) attn_out = P @ v
  k_attnout<<<dim3(N_ / 128, B_ * HEADS_), 256, 0, stream>>>(qkv_post, attn, attn_out);

  // 7) final projection (M=384, K=384, N=36864) -> d_out
  k_conv1x1<<<dim3(N_ / 32, DIM_ / 64, B_), 256, 0, stream>>>(
      w_proj, attn_out, outp, DIM_, DIM_, N_);
}